// GCN_34333968564785
// MI455X (gfx1250) — compile-verified
//
#include <hip/hip_runtime.h>
#include <hip/hip_bf16.h>

typedef __attribute__((ext_vector_type(2))) float v2f;
typedef __attribute__((ext_vector_type(8))) float v8f;

// ---------------------------------------------------------------------------
// K1: init deg (=1.0 for self loop), zero scatter accumulators s and t
// ---------------------------------------------------------------------------
__global__ void gcn_init_kernel(float* __restrict__ deg, float* __restrict__ s,
                                float* __restrict__ t, int n) {
  int i = blockIdx.x * blockDim.x + threadIdx.x;
  if (i < n) { deg[i] = 1.0f; s[i] = 0.0f; t[i] = 0.0f; }
}

// ---------------------------------------------------------------------------
// K2: degree accumulation over edges (dst row of edge_index)
// ---------------------------------------------------------------------------
__global__ void gcn_deg_kernel(const int* __restrict__ ei, float* __restrict__ deg,
                               int E) {
  int e = blockIdx.x * blockDim.x + threadIdx.x;
  if (e < E) atomicAdd(&deg[ei[E + e]], 1.0f);
}

// ---------------------------------------------------------------------------
// K3: dinv = rsqrt(deg) (deg >= 1 always, self loops included); p = dinv * x
// ---------------------------------------------------------------------------
__global__ void gcn_prep_kernel(const float* __restrict__ x,
                                float* __restrict__ deg_dinv,
                                float* __restrict__ p, int n) {
  int i = blockIdx.x * blockDim.x + threadIdx.x;
  if (i < n) {
    float d = rsqrtf(deg_dinv[i]);
    deg_dinv[i] = d;
    p[i] = d * x[i];
  }
}

// ---------------------------------------------------------------------------
// K4/K6: scalar scatter-add over edges: acc[dst] += val[src]
// ---------------------------------------------------------------------------
__global__ void gcn_scatter_kernel(const int* __restrict__ ei,
                                   const float* __restrict__ val,
                                   float* __restrict__ acc, int E) {
  int e = blockIdx.x * blockDim.x + threadIdx.x;
  if (e < E) {
    int sIdx = ei[e];
    int dIdx = ei[E + e];
    atomicAdd(&acc[dIdx], val[sIdx]);
  }
}

// ---------------------------------------------------------------------------
// K5: layer-2 feature transform via f32 WMMA, fused with @Wl row reduction.
//   s_i      = dinv[i] * (s_raw[i] + p[i])              (layer-1 aggregate)
//   h1[i][k] = relu(s_i * W1[k] + b1[k])                (A fragment, in regs)
//   D        = h1_tile(16x16) @ W2(16x16)               (4x V_WMMA_F32_16X16X4_F32)
//   r[i]     = dinv[i] * (D_row(i) . Wl)                (shuffle reduction)
// One wave (32 lanes) per 16-node tile; 8 waves per 256-thread block.
// ISA layouts: A 16x4: lane=M, VGPR v, K = 4*kc + 2*half + v
//              B 4x16 : lane=N, VGPR v, K = 4*kc + 2*half + v
//              D 16x16: lane=N, VGPR v, M = v + 8*half
// ---------------------------------------------------------------------------
__global__ __launch_bounds__(256) void gcn_l2_wmma_kernel(
    const float* __restrict__ s_raw, const float* __restrict__ p,
    const float* __restrict__ dinv, const float* __restrict__ W1,
    const float* __restrict__ b1, const float* __restrict__ W2,
    const float* __restrict__ Wl, float* __restrict__ r, int n_tiles,
    int n_nodes) {
  const int lane = threadIdx.x & 31;
  const int wave = threadIdx.x >> 5;
  const int tile = blockIdx.x * 8 + wave;
  if (tile >= n_tiles) return;  // uniform per wave: EXEC all-1s at WMMA

  const int half = lane >> 4;   // 0: lanes 0-15, 1: lanes 16-31
  const int mn   = lane & 15;   // A-row index / B-column index / D-column index
  const int m0   = tile * 16;

  // layer-1 scalar for this lane's A-row (clamped for safety on ragged tail)
  int node = m0 + mn;
  int nc = node < n_nodes ? node : (n_nodes - 1);
  const float sm = dinv[nc] * (s_raw[nc] + p[nc]);

  v8f acc = {};
#pragma unroll
  for (int kc = 0; kc < 4; ++kc) {
    const int k0 = kc * 4 + half * 2;
    v2f a, b;
    a.x = fmaxf(fmaf(sm, W1[k0 + 0], b1[k0 + 0]), 0.0f);
    a.y = fmaxf(fmaf(sm, W1[k0 + 1], b1[k0 + 1]), 0.0f);
    b.x = W2[(k0 + 0) * 16 + mn];
    b.y = W2[(k0 + 1) * 16 + mn];
    // D = A(16x4) x B(4x16) + C, full f32
    acc = __builtin_amdgcn_wmma_f32_16x16x4_f32(
        /*neg_a=*/false, a, /*neg_b=*/false, b,
        /*c_mod=*/(short)0, acc, /*reuse_a=*/false, /*reuse_b=*/false);
  }

  // Fused projection onto Wl: row-sum of acc[v]*Wl[col] within each 16-lane half
  const float wl = Wl[mn];
  float myval = 0.0f;
#pragma unroll
  for (int v = 0; v < 8; ++v) {
    float tt = acc[v] * wl;
    tt += __shfl_xor(tt, 1, 32);
    tt += __shfl_xor(tt, 2, 32);
    tt += __shfl_xor(tt, 4, 32);
    tt += __shfl_xor(tt, 8, 32);  // masks 1..8 never cross the 16-lane half
    if (mn == v) myval = tt;      // row (v + 8*half) full dot product
  }
  if (mn < 8) {
    int out_node = m0 + mn + 8 * half;
    if (out_node < n_nodes) r[out_node] = dinv[out_node] * myval;
  }
}

// ---------------------------------------------------------------------------
// K7: out[i] = dinv[i] * (r[i] + t[i]) + (b2 @ Wl + bl)
// ---------------------------------------------------------------------------
__global__ void gcn_final_kernel(const float* __restrict__ dinv,
                                 const float* __restrict__ r,
                                 const float* __restrict__ t,
                                 const float* __restrict__ b2,
                                 const float* __restrict__ Wl,
                                 const float* __restrict__ bl,
                                 float* __restrict__ out, int n) {
  int i = blockIdx.x * blockDim.x + threadIdx.x;
  if (i >= n) return;
  float c0 = bl[0];
#pragma unroll
  for (int c = 0; c < 16; ++c) c0 = fmaf(b2[c], Wl[c], c0);
  out[i] = dinv[i] * (r[i] + t[i]) + c0;
}

// ---------------------------------------------------------------------------
// inputs: 0:x[N] 1:edge_index[2*E] 2:W1[16] 3:b1[16] 4:W2[256] 5:b2[16]
//         6:Wl[16] 7:bl[1]      output: [N] float32
// ---------------------------------------------------------------------------
extern "C" void kernel_launch(void* const* d_in, const int* in_sizes, int n_in,
                              void* d_out, int out_size, void* d_ws,
                              size_t ws_size, hipStream_t stream) {
  const float* x  = (const float*)d_in[0];
  const int*   ei = (const int*)d_in[1];
  const float* W1 = (const float*)d_in[2];
  const float* b1 = (const float*)d_in[3];
  const float* W2 = (const float*)d_in[4];
  const float* b2 = (const float*)d_in[5];
  const float* Wl = (const float*)d_in[6];
  const float* bl = (const float*)d_in[7];
  float* out = (float*)d_out;

  const int N = in_sizes[0];
  const int E = in_sizes[1] / 2;

  float* f    = (float*)d_ws;
  float* dinv = f;          // deg then dinv (in place)
  float* p    = f + (size_t)N;
  float* s    = f + 2 * (size_t)N;
  float* r    = f + 3 * (size_t)N;
  float* t    = f + 4 * (size_t)N;

  const int TB = 256;
  const int gN = (N + TB - 1) / TB;
  const int gE = (E + TB - 1) / TB;
  const int n_tiles = (N + 15) / 16;
  const int gT = (n_tiles + 7) / 8;  // 8 waves (tiles) per block

  gcn_init_kernel<<<gN, TB, 0, stream>>>(dinv, s, t, N);
  gcn_deg_kernel<<<gE, TB, 0, stream>>>(ei, dinv, E);
  gcn_prep_kernel<<<gN, TB, 0, stream>>>(x, dinv, p, N);
  gcn_scatter_kernel<<<gE, TB, 0, stream>>>(ei, p, s, E);
  gcn_l2_wmma_kernel<<<gT, TB, 0, stream>>>(s, p, dinv, W1, b1, W2, Wl, r,
                                            n_tiles, N);
  gcn_scatter_kernel<<<gE, TB, 0, stream>>>(ei, r, t, E);
  gcn_final_kernel<<<gN, TB, 0, stream>>>(dinv, r, t, b2, Wl, bl, out, N);
}